// ProductManifoldConv_51135880626689
// MI455X (gfx1250) — compile-verified
//
#include <hip/hip_runtime.h>
#include <hip/hip_bf16.h>

#define EPS 1e-6f

typedef __attribute__((ext_vector_type(2))) float v2f;
typedef __attribute__((ext_vector_type(8))) float v8f;

static constexpr int NN = 2048;   // nodes
static constexpr int MM = 512;    // groups
static constexpr int NITER = 5;

// ---------------------------------------------------------------------------
// Generic fp32 WMMA GEMM:  D[M x N] = sum_k A[k, m] * B[k, n]  (+ bias[n])
// A element = A[k*SAK + m*SAM], B element = B[k*SBK + n*SBN].
// All strides / K compile-time -> loads get immediate offsets, K-stride-1
// operands load as b64 (v2f). One wave per 16x16 tile, K stepped by 4 via
// V_WMMA_F32_16X16X4_F32.
//
// fp32 16x4 A fragment layout (ISA 7.12.2): lane l holds row M = l%16,
// VGPR0 = K = 2*(l>>4), VGPR1 = K = 2*(l>>4)+1. B mirrored. C/D: VGPR r,
// lanes 0-15 -> M=r, lanes 16-31 -> M=r+8, N = l%16.
// ---------------------------------------------------------------------------
template <int SAK, int SAM, int SBK, int SBN, int LDD, int K, bool HAS_BIAS>
__global__ __launch_bounds__(32) void wmma_gemm_f32(
    const float* __restrict__ A, const float* __restrict__ B,
    float* __restrict__ D, const float* __restrict__ bias)
{
  const int lane  = threadIdx.x & 31;
  const int m0    = blockIdx.x * 16;
  const int n0    = blockIdx.y * 16;
  const int row   = m0 + (lane & 15);
  const int col   = n0 + (lane & 15);
  const int khalf = (lane >> 4) * 2;

  v8f acc;
  const float bv = HAS_BIAS ? bias[col] : 0.0f;
#pragma unroll
  for (int r = 0; r < 8; ++r) acc[r] = bv;

  const float* Ap = A + (long)row * SAM + (long)khalf * SAK;
  const float* Bp = B + (long)col * SBN + (long)khalf * SBK;

  constexpr int CH = (K >= 32) ? 32 : K;   // K-chunk per prefetch window
  for (int k0 = 0; k0 < K; k0 += CH) {
    // gfx1250 global_prefetch_b8, one chunk ahead (speculative, OOB-safe)
    __builtin_prefetch(Ap + (long)CH * SAK, 0, 1);
    __builtin_prefetch(Bp + (long)CH * SBK, 0, 1);
#pragma unroll
    for (int kk = 0; kk < CH; kk += 4) {
      v2f a, b;
      if constexpr (SAK == 1) {
        a = *(const v2f*)(Ap + kk);
      } else {
        a.x = Ap[(long)kk * SAK];
        a.y = Ap[(long)kk * SAK + SAK];
      }
      if constexpr (SBK == 1) {
        b = *(const v2f*)(Bp + kk);
      } else {
        b.x = Bp[(long)kk * SBK];
        b.y = Bp[(long)kk * SBK + SBK];
      }
      acc = __builtin_amdgcn_wmma_f32_16x16x4_f32(false, a, false, b,
                                                  (short)0, acc, false, false);
    }
    Ap += (long)CH * SAK;
    Bp += (long)CH * SBK;
  }

  const int rbase = m0 + ((lane >> 4) * 8);
#pragma unroll
  for (int r = 0; r < 8; ++r)
    D[(long)(rbase + r) * LDD + col] = acc[r];
}

// ---------------------------------------------------------------------------
// Elementwise helpers
// ---------------------------------------------------------------------------
__device__ __forceinline__ float vnorm32(const float* v) {
  float s = 0.f;
#pragma unroll
  for (int i = 0; i < 32; ++i) s += v[i] * v[i];
  return sqrtf(s);
}

__device__ __forceinline__ float clip1(float x) {
  return fminf(fmaxf(x, -1.f + EPS), 1.f - EPS);
}

// Ht[g*NN + k] = H[k*MM + g]
__global__ void transpose_kernel(const float* __restrict__ H,
                                 float* __restrict__ Ht) {
  const int k = blockIdx.x * blockDim.x + threadIdx.x;
  const int g = blockIdx.y;
  Ht[(long)g * NN + k] = H[(long)k * MM + g];
}

// x (n x 96) -> U (n x 96): [logmap0(hyp_project(xh)), sphere_project(xs), xe]
__global__ void pre_kernel(const float* __restrict__ x,
                           const float* __restrict__ c_h,
                           float* __restrict__ U) {
  const int i = blockIdx.x * blockDim.x + threadIdx.x;
  if (i >= NN) return;
  const float c  = fabsf(*c_h) + EPS;
  const float sc = sqrtf(c);
  const float* xr = x + (long)i * 96;
  float* Ur = U + (long)i * 96;

  float h[32];
#pragma unroll
  for (int j = 0; j < 32; ++j) h[j] = xr[j];
  const float max_norm = (1.0f - 1e-5f) / sc;
  float hn = fmaxf(vnorm32(h), EPS);
  if (hn > max_norm) {
    const float r = max_norm / hn;
#pragma unroll
    for (int j = 0; j < 32; ++j) h[j] *= r;
  }
  float n2 = fmaxf(vnorm32(h), EPS);
  float f = atanhf(clip1(sc * n2)) / (sc * n2);
#pragma unroll
  for (int j = 0; j < 32; ++j) Ur[j] = f * h[j];

  float s[32];
#pragma unroll
  for (int j = 0; j < 32; ++j) s[j] = xr[32 + j];
  const float inv = 1.0f / fmaxf(vnorm32(s), EPS);
#pragma unroll
  for (int j = 0; j < 32; ++j) Ur[32 + j] = s[j] * inv;

#pragma unroll
  for (int j = 0; j < 32; ++j) Ur[64 + j] = xr[64 + j];
}

// Y (n x 96) -> Xh, Xs, Xe (n x 32 each)
__global__ void post_kernel(const float* __restrict__ Y,
                            const float* __restrict__ c_h,
                            float* __restrict__ Xh, float* __restrict__ Xs,
                            float* __restrict__ Xe) {
  const int i = blockIdx.x * blockDim.x + threadIdx.x;
  if (i >= NN) return;
  const float c  = fabsf(*c_h) + EPS;
  const float sc = sqrtf(c);
  const float* yr = Y + (long)i * 96;

  float v[32];
#pragma unroll
  for (int j = 0; j < 32; ++j) v[j] = yr[j];
  float n = fmaxf(vnorm32(v), EPS);
  float f = tanhf(sc * n) / (sc * n);
#pragma unroll
  for (int j = 0; j < 32; ++j) v[j] *= f;
  const float max_norm = (1.0f - 1e-5f) / sc;
  float hn = fmaxf(vnorm32(v), EPS);
  if (hn > max_norm) {
    const float r = max_norm / hn;
#pragma unroll
    for (int j = 0; j < 32; ++j) v[j] *= r;
  }
#pragma unroll
  for (int j = 0; j < 32; ++j) Xh[(long)i * 32 + j] = v[j];

  float s[32];
#pragma unroll
  for (int j = 0; j < 32; ++j) s[j] = yr[32 + j];
  const float inv = 1.0f / fmaxf(vnorm32(s), EPS);
#pragma unroll
  for (int j = 0; j < 32; ++j) Xs[(long)i * 32 + j] = s[j] * inv;

#pragma unroll
  for (int j = 0; j < 32; ++j) Xe[(long)i * 32 + j] = yr[64 + j];
}

// points (npts x 32) -> Ppad (npts x 48): [lam*p, lam-1, zeros]
__global__ void gyro_lam_kernel(const float* __restrict__ P,
                                const float* __restrict__ c_h,
                                float* __restrict__ Ppad, int npts) {
  const int i = blockIdx.x * blockDim.x + threadIdx.x;
  if (i >= npts) return;
  const float c = fabsf(*c_h) + EPS;
  const float* p = P + (long)i * 32;
  float sq = 0.f;
#pragma unroll
  for (int j = 0; j < 32; ++j) sq += p[j] * p[j];
  const float lam = 2.0f / fmaxf(1.0f - c * sq, EPS);
  float* o = Ppad + (long)i * 48;
#pragma unroll
  for (int j = 0; j < 32; ++j) o[j] = lam * p[j];
  o[32] = lam - 1.0f;
#pragma unroll
  for (int j = 33; j < 48; ++j) o[j] = 0.0f;
}

// Epad (G x 48: [num, den, pad]) -> out (G x 32): Mobius gyromidpoint close
__global__ void gyro_post_kernel(const float* __restrict__ Epad,
                                 const float* __restrict__ c_h,
                                 float* __restrict__ out, int G) {
  const int g = blockIdx.x * blockDim.x + threadIdx.x;
  if (g >= G) return;
  const float c  = fabsf(*c_h) + EPS;
  const float sc = sqrtf(c);
  const float* r = Epad + (long)g * 48;
  const float den = fmaxf(r[32], EPS);
  float y[32];
#pragma unroll
  for (int j = 0; j < 32; ++j) y[j] = r[j] / den;
  const float yn = fmaxf(vnorm32(y), EPS);
  const float f = tanhf(0.5f * atanhf(clip1(sc * yn))) / (sc * yn);
#pragma unroll
  for (int j = 0; j < 32; ++j) out[(long)g * 32 + j] = f * y[j];
}

__global__ void normalize_kernel(float* __restrict__ buf, int G) {
  const int g = blockIdx.x * blockDim.x + threadIdx.x;
  if (g >= G) return;
  float* r = buf + (long)g * 32;
  float v[32];
#pragma unroll
  for (int j = 0; j < 32; ++j) v[j] = r[j];
  const float inv = 1.0f / fmaxf(vnorm32(v), EPS);
#pragma unroll
  for (int j = 0; j < 32; ++j) r[j] = v[j] * inv;
}

__global__ void scale_rows_kernel(float* __restrict__ buf,
                                  const float* __restrict__ deg, int G) {
  const int g = blockIdx.x * blockDim.x + threadIdx.x;
  if (g >= G) return;
  const float inv = 1.0f / fmaxf(deg[g], EPS);
#pragma unroll
  for (int j = 0; j < 32; ++j) buf[(long)g * 32 + j] *= inv;
}

// out[i] = sum_{j<cnt} src[i*so + j*si]
__global__ void rowsum_kernel(const float* __restrict__ src, int so, int si,
                              int cnt, float* __restrict__ out, int nout) {
  const int i = blockIdx.x * blockDim.x + threadIdx.x;
  if (i >= nout) return;
  const float* p = src + (long)i * so;
  float s = 0.f;
  for (int j = 0; j < cnt; ++j) s += p[(long)j * si];
  out[i] = s;
}

// dots (G x Kn) -> C, CD. w(k,g) = Hm[k*hk + g*hg] / max(deg[g], eps)
// coef = w * acos(dot_c) / ||y - dot_c*mu|| with unit mu, y.
__global__ void coef_kernel(const float* __restrict__ dots,
                            const float* __restrict__ Hm, int hk, int hg,
                            const float* __restrict__ deg,
                            float* __restrict__ C, float* __restrict__ CD,
                            int Kn) {
  const int k = blockIdx.x * blockDim.x + threadIdx.x;
  const int g = blockIdx.y;
  if (k >= Kn) return;
  const long idx = (long)g * Kn + k;
  const float d  = dots[idx];
  const float dc = clip1(d);
  const float theta = acosf(dc);
  const float dn2 = fmaxf(1.0f - 2.0f * dc * d + dc * dc, 0.0f);
  const float dirn = fmaxf(sqrtf(dn2), EPS);
  const float w = Hm[(long)k * hk + (long)g * hg] / fmaxf(deg[g], EPS);
  const float cf = w * theta / dirn;
  C[idx] = cf;
  CD[idx] = cf * dc;
}

// mu <- sphere_project(sphere_exp(mu, s1 - s2*mu))
__global__ void sphere_update_kernel(float* __restrict__ mu,
                                     const float* __restrict__ s1,
                                     const float* __restrict__ s2, int G) {
  const int g = blockIdx.x * blockDim.x + threadIdx.x;
  if (g >= G) return;
  float m[32], v[32], r[32];
  const float s2g = s2[g];
#pragma unroll
  for (int j = 0; j < 32; ++j) {
    m[j] = mu[(long)g * 32 + j];
    v[j] = s1[(long)g * 32 + j] - s2g * m[j];
  }
  const float vn  = vnorm32(v);
  const float vns = fmaxf(vn, EPS);
  const float cv = cosf(vn);
  const float sv = sinf(vn) / vns;
#pragma unroll
  for (int j = 0; j < 32; ++j) r[j] = cv * m[j] + sv * v[j];
  const float inv = 1.0f / fmaxf(vnorm32(r), EPS);
#pragma unroll
  for (int j = 0; j < 32; ++j) mu[(long)g * 32 + j] = r[j] * inv;
}

// concat [hyp_project(outh), sphere_project(mus), oute/deg_v] -> out (n x 96)
__global__ void final_kernel(const float* __restrict__ outh,
                             const float* __restrict__ mus,
                             const float* __restrict__ oute,
                             const float* __restrict__ deg_v,
                             const float* __restrict__ c_h,
                             float* __restrict__ out) {
  const int i = blockIdx.x * blockDim.x + threadIdx.x;
  if (i >= NN) return;
  const float c  = fabsf(*c_h) + EPS;
  const float sc = sqrtf(c);
  float* o = out + (long)i * 96;

  float h[32];
#pragma unroll
  for (int j = 0; j < 32; ++j) h[j] = outh[(long)i * 32 + j];
  const float max_norm = (1.0f - 1e-5f) / sc;
  float hn = fmaxf(vnorm32(h), EPS);
  float rr = (hn > max_norm) ? (max_norm / hn) : 1.0f;
#pragma unroll
  for (int j = 0; j < 32; ++j) o[j] = h[j] * rr;

  float s[32];
#pragma unroll
  for (int j = 0; j < 32; ++j) s[j] = mus[(long)i * 32 + j];
  const float inv = 1.0f / fmaxf(vnorm32(s), EPS);
#pragma unroll
  for (int j = 0; j < 32; ++j) o[32 + j] = s[j] * inv;

  const float invd = 1.0f / fmaxf(deg_v[i], EPS);
#pragma unroll
  for (int j = 0; j < 32; ++j) o[64 + j] = oute[(long)i * 32 + j] * invd;
}

// ---------------------------------------------------------------------------
extern "C" void kernel_launch(void* const* d_in, const int* in_sizes, int n_in,
                              void* d_out, int out_size, void* d_ws,
                              size_t ws_size, hipStream_t stream) {
  (void)in_sizes; (void)n_in; (void)out_size; (void)ws_size;
  const float* x   = (const float*)d_in[0];
  const float* H   = (const float*)d_in[1];
  const float* W_h = (const float*)d_in[2];
  const float* b_h = (const float*)d_in[3];
  const float* W_s = (const float*)d_in[4];
  const float* b_s = (const float*)d_in[5];
  const float* W_e = (const float*)d_in[6];
  const float* b_e = (const float*)d_in[7];
  const float* c_h = (const float*)d_in[8];
  // c_s (d_in[9]) is unused by the reference math.

  float* ws = (float*)d_ws;
  auto alloc = [&](long nflt) { float* p = ws; ws += nflt; return p; };
  float* U     = alloc((long)NN * 96);
  float* Y     = alloc((long)NN * 96);
  float* Xh    = alloc((long)NN * 32);
  float* Xs    = alloc((long)NN * 32);
  float* Xe    = alloc((long)NN * 32);
  float* Ppad  = alloc((long)NN * 48);   // reused by both stages
  float* Epad  = alloc((long)NN * 48);   // reused by both stages
  float* eh    = alloc((long)MM * 32);
  float* es    = alloc((long)MM * 32);   // becomes e_s after iterations
  float* ee    = alloc((long)MM * 32);
  float* deg_e = alloc(MM);
  float* deg_v = alloc(NN);
  float* Ht    = alloc((long)MM * NN);   // H transposed: Ht[g*NN + k]
  float* dots  = alloc((long)NN * MM);
  float* Cm    = alloc((long)NN * MM);
  float* CDm   = alloc((long)NN * MM);
  float* s1b   = alloc((long)NN * 32);
  float* s2b   = alloc(NN);
  float* outh  = alloc((long)NN * 32);
  float* mu2   = alloc((long)NN * 32);
  float* oute  = alloc((long)NN * 32);

  const dim3 B256(256);
  const dim3 W32(32);
  const dim3 GN((NN + 255) / 256);
  const dim3 GM((MM + 255) / 256);

  // ---- H transpose (makes every GEMM A-operand K-contiguous) ----
  transpose_kernel<<<dim3(NN / 256, MM), B256, 0, stream>>>(H, Ht);

  // ---- node transforms ----
  pre_kernel<<<GN, B256, 0, stream>>>(x, c_h, U);
  wmma_gemm_f32<1, 96, 1, 32, 96, 32, true>
      <<<dim3(NN / 16, 2), W32, 0, stream>>>(U + 0, W_h, Y + 0, b_h);
  wmma_gemm_f32<1, 96, 1, 32, 96, 32, true>
      <<<dim3(NN / 16, 2), W32, 0, stream>>>(U + 32, W_s, Y + 32, b_s);
  wmma_gemm_f32<1, 96, 1, 32, 96, 32, true>
      <<<dim3(NN / 16, 2), W32, 0, stream>>>(U + 64, W_e, Y + 64, b_e);
  post_kernel<<<GN, B256, 0, stream>>>(Y, c_h, Xh, Xs, Xe);

  // ---- degrees ----
  rowsum_kernel<<<GM, B256, 0, stream>>>(Ht, NN, 1, NN, deg_e, MM);
  rowsum_kernel<<<GN, B256, 0, stream>>>(H, MM, 1, MM, deg_v, NN);

  // ---- stage 1: nodes -> groups (A = Ht: SAK=1, SAM=NN, K=NN) ----
  gyro_lam_kernel<<<GN, B256, 0, stream>>>(Xh, c_h, Ppad, NN);
  wmma_gemm_f32<1, NN, 48, 1, 48, NN, false>
      <<<dim3(MM / 16, 3), W32, 0, stream>>>(Ht, Ppad, Epad, nullptr);
  gyro_post_kernel<<<GM, B256, 0, stream>>>(Epad, c_h, eh, MM);

  wmma_gemm_f32<1, NN, 32, 1, 32, NN, false>
      <<<dim3(MM / 16, 2), W32, 0, stream>>>(Ht, Xe, ee, nullptr);
  scale_rows_kernel<<<GM, B256, 0, stream>>>(ee, deg_e, MM);

  wmma_gemm_f32<1, NN, 32, 1, 32, NN, false>
      <<<dim3(MM / 16, 2), W32, 0, stream>>>(Ht, Xs, es, nullptr);
  normalize_kernel<<<GM, B256, 0, stream>>>(es, MM);
  for (int it = 0; it < NITER; ++it) {
    wmma_gemm_f32<1, 32, 1, 32, NN, 32, false>
        <<<dim3(MM / 16, NN / 16), W32, 0, stream>>>(es, Xs, dots, nullptr);
    coef_kernel<<<dim3(NN / 256, MM), B256, 0, stream>>>(dots, Ht, 1, NN,
                                                         deg_e, Cm, CDm, NN);
    rowsum_kernel<<<GM, B256, 0, stream>>>(CDm, NN, 1, NN, s2b, MM);
    wmma_gemm_f32<1, NN, 32, 1, 32, NN, false>
        <<<dim3(MM / 16, 2), W32, 0, stream>>>(Cm, Xs, s1b, nullptr);
    sphere_update_kernel<<<GM, B256, 0, stream>>>(es, s1b, s2b, MM);
  }

  // ---- stage 2: groups -> nodes (A = H: SAK=1, SAM=MM, K=MM) ----
  gyro_lam_kernel<<<GM, B256, 0, stream>>>(eh, c_h, Ppad, MM);
  wmma_gemm_f32<1, MM, 48, 1, 48, MM, false>
      <<<dim3(NN / 16, 3), W32, 0, stream>>>(H, Ppad, Epad, nullptr);
  gyro_post_kernel<<<GN, B256, 0, stream>>>(Epad, c_h, outh, NN);

  wmma_gemm_f32<1, MM, 32, 1, 32, MM, false>
      <<<dim3(NN / 16, 2), W32, 0, stream>>>(H, ee, oute, nullptr);

  wmma_gemm_f32<1, MM, 32, 1, 32, MM, false>
      <<<dim3(NN / 16, 2), W32, 0, stream>>>(H, es, mu2, nullptr);
  normalize_kernel<<<GN, B256, 0, stream>>>(mu2, NN);
  for (int it = 0; it < NITER; ++it) {
    wmma_gemm_f32<1, 32, 1, 32, MM, 32, false>
        <<<dim3(NN / 16, MM / 16), W32, 0, stream>>>(mu2, es, dots, nullptr);
    coef_kernel<<<dim3(MM / 256, NN), B256, 0, stream>>>(dots, H, 1, MM,
                                                         deg_v, Cm, CDm, MM);
    rowsum_kernel<<<GN, B256, 0, stream>>>(CDm, MM, 1, MM, s2b, NN);
    wmma_gemm_f32<1, MM, 32, 1, 32, MM, false>
        <<<dim3(NN / 16, 2), W32, 0, stream>>>(Cm, es, s1b, nullptr);
    sphere_update_kernel<<<GN, B256, 0, stream>>>(mu2, s1b, s2b, NN);
  }

  final_kernel<<<GN, B256, 0, stream>>>(outh, mu2, oute, deg_v, c_h,
                                        (float*)d_out);
}